// CSG2A_net_5712306503848
// MI455X (gfx1250) — compile-verified
//
#include <hip/hip_runtime.h>
#include <hip/hip_bf16.h>

#define GEX 978
#define KPAD 980
#define HDIM 128
#define NAT 50
#define FEAT 34
#define CCEH 64
#define BATCH 64

typedef __attribute__((ext_vector_type(2))) float v2f;
typedef __attribute__((ext_vector_type(8))) float v8f;

// ---------------- CCE: relu(nf@W1), dist-weighted MP, masked mean pool ----
__global__ __launch_bounds__(256) void k_cce(const float* __restrict__ nf,
                                             const float* __restrict__ mask,
                                             const float* __restrict__ adj,
                                             const float* __restrict__ dist,
                                             const float* __restrict__ W1,
                                             float* __restrict__ pooled) {
    int b = blockIdx.x, tid = threadIdx.x;
    __shared__ float W1s[FEAT * CCEH];
    __shared__ float hs[NAT * CCEH];
    __shared__ float wm[NAT];
    for (int i = tid; i < FEAT * CCEH; i += 256) W1s[i] = W1[i];
    __syncthreads();
    for (int i = tid; i < NAT * CCEH; i += 256) {
        int n = i / CCEH, d = i % CCEH;
        const float* row = nf + (b * NAT + n) * FEAT;
        float s = 0.f;
        #pragma unroll
        for (int f = 0; f < FEAT; f++) s += row[f] * W1s[f * CCEH + d];
        hs[i] = fmaxf(s, 0.f);
    }
    if (tid < NAT) {
        float s = 0.f;
        for (int n = 0; n < NAT; n++) {
            int idx = (b * NAT + n) * NAT + tid;
            s += mask[b * NAT + n] * adj[idx] * expf(-dist[idx]);
        }
        wm[tid] = s;
    }
    __syncthreads();
    if (tid < CCEH) {
        float msum = 0.f;
        for (int n = 0; n < NAT; n++) msum += mask[b * NAT + n];
        msum = fmaxf(msum, 1.f);
        float s = 0.f;
        for (int m = 0; m < NAT; m++) s += wm[m] * hs[m * CCEH + tid];
        pooled[b * CCEH + tid] = s / msum;
    }
}

// ---------------- comp = pooled@W2 + dose@w_dose + time@w_time -----------
__global__ __launch_bounds__(256) void k_comp(const float* __restrict__ pooled,
                                              const float* __restrict__ W2,
                                              const float* __restrict__ dose,
                                              const float* __restrict__ timev,
                                              const float* __restrict__ w_dose,
                                              const float* __restrict__ w_time,
                                              float* __restrict__ comp) {
    int idx = blockIdx.x * 256 + threadIdx.x;
    if (idx >= BATCH * GEX) return;
    int b = idx / GEX, g = idx % GEX;
    float s = dose[b] * w_dose[g] + timev[b] * w_time[g];
    #pragma unroll 8
    for (int d = 0; d < CCEH; d++) s += pooled[b * CCEH + d] * W2[d * GEX + g];
    comp[idx] = s;
}

// ---------------- S[b,:] = sum_g b_gex[b,g]*w_gex[g,:] + comp[b,g]*w_comp[g,:]
__global__ __launch_bounds__(128) void k_S(const float* __restrict__ b_gex,
                                           const float* __restrict__ comp,
                                           const float* __restrict__ w_gex,
                                           const float* __restrict__ w_comp,
                                           float* __restrict__ S) {
    int b = blockIdx.x, h = threadIdx.x;
    float s = 0.f;
    for (int g = 0; g < GEX; g++) {
        float x = b_gex[b * GEX + g], c = comp[b * GEX + g];
        s += x * w_gex[g * HDIM + h] + c * w_comp[g * HDIM + h];
    }
    S[b * HDIM + h] = s;
}

// ---------------- ppi row sums -------------------------------------------
__global__ __launch_bounds__(256) void k_ppirow(const float* __restrict__ ppi,
                                                float* __restrict__ rowsum) {
    int g = blockIdx.x, tid = threadIdx.x;
    __shared__ float red[256];
    float s = 0.f;
    for (int k = tid; k < GEX; k += 256) s += ppi[g * GEX + k];
    red[tid] = s;
    __syncthreads();
    for (int o = 128; o > 0; o >>= 1) {
        if (tid < o) red[tid] += red[tid + o];
        __syncthreads();
    }
    if (tid == 0) rowsum[g] = red[0];
}

// ---------------- pred0[b,g] = x*( (x*(wg[g].S[b]) + c*(wc[g].S[b]))/sqrt(H) + ppirow[g] )
__global__ __launch_bounds__(256) void k_pred0(const float* __restrict__ b_gex,
                                               const float* __restrict__ comp,
                                               const float* __restrict__ w_gex,
                                               const float* __restrict__ w_comp,
                                               const float* __restrict__ S,
                                               const float* __restrict__ ppirow,
                                               float* __restrict__ pred0) {
    __shared__ float Ssh[BATCH * HDIM];       // 32 KB
    __shared__ float wg[4][HDIM], wc[4][HDIM];
    int g0 = blockIdx.x * 4, tid = threadIdx.x;
    for (int i = tid; i < BATCH * HDIM; i += 256) Ssh[i] = S[i];
    for (int i = tid; i < 4 * HDIM; i += 256) {
        int gl = i / HDIM, hh = i % HDIM, g = g0 + gl;
        wg[gl][hh] = (g < GEX) ? w_gex[g * HDIM + hh] : 0.f;
        wc[gl][hh] = (g < GEX) ? w_comp[g * HDIM + hh] : 0.f;
    }
    __syncthreads();
    int b = tid >> 2, gl = tid & 3, g = g0 + gl;
    if (g < GEX) {
        float d1 = 0.f, d2 = 0.f;
        #pragma unroll 8
        for (int hh = 0; hh < HDIM; hh++) {
            float sv = Ssh[b * HDIM + hh];
            d1 += sv * wg[gl][hh];
            d2 += sv * wc[gl][hh];
        }
        float x = b_gex[b * GEX + g], c = comp[b * GEX + g];
        float row = (x * d1 + c * d2) * 0.08838834764831845f + ppirow[g]; // 1/sqrt(128)
        pred0[b * GEX + g] = x * row;
    }
}

// ---------------- LayerNorm + ReLU -> padded [B][KPAD] -------------------
__global__ __launch_bounds__(256) void k_ln(const float* __restrict__ pred0,
                                            const float* __restrict__ gamma,
                                            const float* __restrict__ beta,
                                            float* __restrict__ xlnP) {
    int b = blockIdx.x, tid = threadIdx.x;
    __shared__ float red[256];
    __shared__ float mu_s, rstd_s;
    float s = 0.f;
    for (int g = tid; g < GEX; g += 256) s += pred0[b * GEX + g];
    red[tid] = s;
    __syncthreads();
    for (int o = 128; o > 0; o >>= 1) { if (tid < o) red[tid] += red[tid + o]; __syncthreads(); }
    if (tid == 0) mu_s = red[0] / (float)GEX;
    __syncthreads();
    float mu = mu_s, v = 0.f;
    for (int g = tid; g < GEX; g += 256) { float d = pred0[b * GEX + g] - mu; v += d * d; }
    red[tid] = v;
    __syncthreads();
    for (int o = 128; o > 0; o >>= 1) { if (tid < o) red[tid] += red[tid + o]; __syncthreads(); }
    if (tid == 0) rstd_s = rsqrtf(red[0] / (float)GEX + 1e-5f);
    __syncthreads();
    float rstd = rstd_s;
    for (int g = tid; g < KPAD; g += 256) {
        float y = 0.f;
        if (g < GEX)
            y = fmaxf((pred0[b * GEX + g] - mu) * rstd * gamma[g] + beta[g], 0.f);
        xlnP[b * KPAD + g] = y;   // K-padding zeros at 978,979
    }
}

// ---------------- W_ff transpose -> WffT[n][k], k-padded to KPAD ---------
__global__ __launch_bounds__(256) void k_transpose(const float* __restrict__ Wff,
                                                   float* __restrict__ WffT) {
    __shared__ float tile[32][33];
    int kb = blockIdx.x * 32;   // source row block (k)
    int nb = blockIdx.y * 32;   // source col block (n)
    int tx = threadIdx.x & 31, ty = threadIdx.x >> 5; // 32x8
    for (int i = ty; i < 32; i += 8) {
        int k = kb + i, n = nb + tx;
        tile[i][tx] = (k < GEX && n < GEX) ? Wff[k * GEX + n] : 0.f;
    }
    __syncthreads();
    for (int i = ty; i < 32; i += 8) {
        int n = nb + i, k = kb + tx;
        if (n < GEX && k < KPAD) WffT[n * KPAD + k] = tile[tx][i];
    }
}

// ---------------- out = xln @ W_ff via V_WMMA_F32_16X16X4_F32 ------------
// One wave per 16x16 output tile; branch-free K loop, b64 fragment loads.
// A 16x4 layout: lanes 0-15 -> M=lane, K=k,k+1 ; lanes 16-31 -> M=lane-16, K=k+2,k+3.
// B 4x16: same K split per lane group, N = lane&15. Lanes with n>=978 clamp their
// B row pointer (garbage only feeds never-stored D columns). K-pad rows of A are
// zero, so bogus B rows at k=978,979 contribute nothing.
__global__ __launch_bounds__(256) void k_gemm_wmma(const float* __restrict__ xlnP,
                                                   const float* __restrict__ WffT,
                                                   float* __restrict__ out) {
    const int NT = (GEX + 15) / 16; // 62
    int wave = (blockIdx.x * 256 + threadIdx.x) >> 5;
    int lane = threadIdx.x & 31;
    int mt = wave / NT;
    int nt = wave % NT;
    if (mt >= BATCH / 16) return;   // wave-uniform (never taken with exact grid)
    int lg   = lane & 15;
    int koff = (lane >= 16) ? 2 : 0;
    int m = mt * 16 + lg;
    int n = nt * 16 + lg;
    int nc = (n < GEX) ? n : (GEX - 1);

    const float* ap = xlnP + m * KPAD + koff;
    const float* bp = WffT + nc * KPAD + koff;

    v8f acc = {};
    #pragma unroll 5
    for (int k = 0; k < KPAD; k += 4) {   // 245 branch-free iterations
        v2f a  = *(const v2f*)ap;
        v2f bm = *(const v2f*)bp;
        ap += 4; bp += 4;
        acc = __builtin_amdgcn_wmma_f32_16x16x4_f32(
            false, a, false, bm, (short)0, acc, false, false);
    }

    bool nok = n < GEX;
    #pragma unroll
    for (int v = 0; v < 8; v++) {
        int row = mt * 16 + v + ((lane >= 16) ? 8 : 0);
        if (nok) out[row * GEX + n] = acc[v];
    }
}

extern "C" void kernel_launch(void* const* d_in, const int* in_sizes, int n_in,
                              void* d_out, int out_size, void* d_ws, size_t ws_size,
                              hipStream_t stream) {
    const float* b_gex    = (const float*)d_in[0];
    const float* node_feat= (const float*)d_in[1];
    const float* mask     = (const float*)d_in[2];
    const float* adj      = (const float*)d_in[3];
    const float* dist     = (const float*)d_in[4];
    const float* dose     = (const float*)d_in[5];
    const float* timev    = (const float*)d_in[6];
    const float* ppi      = (const float*)d_in[7];
    const float* w_gex    = (const float*)d_in[8];
    const float* w_comp   = (const float*)d_in[9];
    const float* W1       = (const float*)d_in[10];
    const float* W2       = (const float*)d_in[11];
    const float* w_dose   = (const float*)d_in[12];
    const float* w_time   = (const float*)d_in[13];
    const float* gamma    = (const float*)d_in[14];
    const float* beta     = (const float*)d_in[15];
    const float* Wff      = (const float*)d_in[16];

    float* out  = (float*)d_out;             // [B,G] pred_gex
    float* comp = out + BATCH * GEX;         // [B,G] comp (second output)

    float* ws     = (float*)d_ws;
    float* pooled = ws;                              // 64*64        = 4096
    float* S      = pooled + BATCH * CCEH;           // 64*128       = 8192
    float* ppirow = S + BATCH * HDIM;                // 978 (pad 1024)
    float* pred0  = ppirow + 1024;                   // 64*978
    float* xlnP   = pred0 + BATCH * GEX;             // 64*980 (K-padded)
    float* WffT   = xlnP + BATCH * KPAD;             // 978*980 transposed, padded

    // W_ff transpose is independent of the pipeline; launch first.
    dim3 tg((KPAD + 31) / 32, (GEX + 31) / 32);
    k_transpose<<<tg, 256, 0, stream>>>(Wff, WffT);

    k_cce<<<BATCH, 256, 0, stream>>>(node_feat, mask, adj, dist, W1, pooled);
    k_comp<<<(BATCH * GEX + 255) / 256, 256, 0, stream>>>(pooled, W2, dose, timev,
                                                          w_dose, w_time, comp);
    k_S<<<BATCH, HDIM, 0, stream>>>(b_gex, comp, w_gex, w_comp, S);
    k_ppirow<<<GEX, 256, 0, stream>>>(ppi, ppirow);
    k_pred0<<<(GEX + 3) / 4, 256, 0, stream>>>(b_gex, comp, w_gex, w_comp, S, ppirow, pred0);
    k_ln<<<BATCH, 256, 0, stream>>>(pred0, gamma, beta, xlnP);
    // 4 M-tiles * 62 N-tiles = 248 waves = 31 blocks of 8 waves
    k_gemm_wmma<<<31, 256, 0, stream>>>(xlnP, WffT, out);
}